// GCN_2_5411658793404
// MI455X (gfx1250) — compile-verified
//
#include <hip/hip_runtime.h>
#include <hip/hip_bf16.h>
#include <math.h>

// ---------------------------------------------------------------------------
// SemGCN forward for MI455X (gfx1250, wave32).
// GEMMs on the f32 matrix pipe via V_WMMA_F32_16X16X4_F32, with double-
// buffered async global->LDS staging (ASYNCcnt) when available.
// ---------------------------------------------------------------------------

typedef float v2f __attribute__((ext_vector_type(2)));
typedef float v8f __attribute__((ext_vector_type(8)));

__device__ __forceinline__ v8f wmma4(v2f a, v2f b, v8f c) {
    // (neg_a, A, neg_b, B, c_mod, C, reuse_a, reuse_b)
    return __builtin_amdgcn_wmma_f32_16x16x4_f32(false, a, false, b, (short)0, c,
                                                 false, false);
}

#ifndef __has_builtin
#define __has_builtin(x) 0
#endif

#if __has_builtin(__builtin_amdgcn_global_load_async_to_lds_b32)
#define ASYNC_STAGE 1
typedef __attribute__((address_space(1))) int gint_t;   // global int*
typedef __attribute__((address_space(3))) int lint_t;   // LDS int*
#else
#define ASYNC_STAGE 0
#endif

__device__ __forceinline__ void stage_b32(const float* __restrict__ g,
                                          float* __restrict__ l) {
#if ASYNC_STAGE
    __builtin_amdgcn_global_load_async_to_lds_b32(
        (gint_t*)(float*)g, (lint_t*)l, 0, 0);
#else
    *l = *g;
#endif
}

__device__ __forceinline__ void wait_async0() {
#if ASYNC_STAGE
#if __has_builtin(__builtin_amdgcn_s_wait_asynccnt)
    __builtin_amdgcn_s_wait_asynccnt(0);
#else
    asm volatile("s_wait_asynccnt 0" ::: "memory");
#endif
#endif
}

#define GBM 256
#define GBN 64
#define GKT 16
#define LSTR 18              // even LDS stride: 8B-aligned fragments, no conflicts
#define STATS_BLOCKS 256

// ---------------------------------------------------------------------------
// GEMM: H[R, 2F] = X[R, K] @ [W0 | W1]   (W0, W1 each [K, F] row-major)
// Requires K % 16 == 0, R % 256 == 0, (2F) % 64 == 0 (true for all layers).
// Block = 256 threads = 8 waves, tile 256x64; each wave: 32 rows x 64 cols
// (2x4 v8f accumulators, 8 WMMA per k-step). Double-buffered LDS staging.
// ---------------------------------------------------------------------------
__global__ __launch_bounds__(256) void gemm_wmma(
    const float* __restrict__ X, const float* __restrict__ W0,
    const float* __restrict__ W1, float* __restrict__ H,
    int R, int K, int F)
{
    const int N = 2 * F;
    __shared__ float Xs[2][GBM][LSTR];   // 2*256*18*4 = 36 KB
    __shared__ float Wt[2][GBN][LSTR];   // transposed W tile: Wt[n][k], 9 KB

    const int t    = threadIdx.x;
    const int lane = t & 31;
    const int wid  = t >> 5;
    const int tileM = blockIdx.y * GBM;
    const int tileN = blockIdx.x * GBN;

    const int half = lane >> 4;   // lanes 16-31 hold K+2,K+3 of each fragment
    const int lm   = lane & 15;
    const int m0   = wid * 32 + lm;

    v8f acc[2][4] = {};

    // stage one 256x16 X slab + 16x64 W slab (transposed) into buffer `buf`
    auto stage = [&](int buf, int k0) {
        #pragma unroll
        for (int j = 0; j < 16; j++) {          // 4096 dwords / 256 threads
            int i = t + j * 256;
            int r = i >> 4, c = i & 15;
            stage_b32(X + (size_t)(tileM + r) * K + (k0 + c), &Xs[buf][r][c]);
        }
        #pragma unroll
        for (int j = 0; j < 4; j++) {           // 1024 dwords / 256 threads
            int i = t + j * 256;
            int kr = i >> 6, n = i & 63;
            int gn = tileN + n;
            const float* g = (gn < F) ? &W0[(size_t)(k0 + kr) * F + gn]
                                      : &W1[(size_t)(k0 + kr) * F + (gn - F)];
            stage_b32(g, &Wt[buf][n][kr]);
        }
    };

    stage(0, 0);
    wait_async0();
    __syncthreads();

    int buf = 0;
    for (int k0 = 0; k0 < K; k0 += GKT) {
        if (k0 + GKT < K) stage(buf ^ 1, k0 + GKT);   // overlap with compute

        #pragma unroll
        for (int kk = 0; kk < GKT; kk += 4) {
            const int ka = kk + 2 * half;
            v2f a0 = *(const v2f*)&Xs[buf][m0][ka];
            v2f a1 = *(const v2f*)&Xs[buf][m0 + 16][ka];
            v2f b0 = *(const v2f*)&Wt[buf][lm][ka];
            v2f b1 = *(const v2f*)&Wt[buf][lm + 16][ka];
            v2f b2 = *(const v2f*)&Wt[buf][lm + 32][ka];
            v2f b3 = *(const v2f*)&Wt[buf][lm + 48][ka];
            acc[0][0] = wmma4(a0, b0, acc[0][0]);
            acc[1][0] = wmma4(a1, b0, acc[1][0]);
            acc[0][1] = wmma4(a0, b1, acc[0][1]);
            acc[1][1] = wmma4(a1, b1, acc[1][1]);
            acc[0][2] = wmma4(a0, b2, acc[0][2]);
            acc[1][2] = wmma4(a1, b2, acc[1][2]);
            acc[0][3] = wmma4(a0, b3, acc[0][3]);
            acc[1][3] = wmma4(a1, b3, acc[1][3]);
        }
        wait_async0();
        __syncthreads();
        buf ^= 1;
    }

    // store: C/D layout: VGPR r -> row r (lanes 0-15) / row r+8 (lanes 16-31)
    #pragma unroll
    for (int im = 0; im < 2; im++) {
        const int rowBase = tileM + wid * 32 + im * 16 + 8 * half;
        #pragma unroll
        for (int q = 0; q < 4; q++) {
            const int col = tileN + q * 16 + lm;
            #pragma unroll
            for (int r = 0; r < 8; r++)
                H[(size_t)(rowBase + r) * N + col] = acc[im][q][r];
        }
    }
}

// ---------------------------------------------------------------------------
// K=2 projection (input layers): H[row, col] = x0*Wc(0,col) + x1*Wc(1,col)
// ---------------------------------------------------------------------------
__global__ __launch_bounds__(256) void gemm_k2(
    const float* __restrict__ X, const float* __restrict__ W0,
    const float* __restrict__ W1, float* __restrict__ H, int R, int F)
{
    const int N = 2 * F;
    size_t gid = (size_t)blockIdx.x * 256 + threadIdx.x;
    if (gid >= (size_t)R * N) return;
    int col = (int)(gid % N);
    size_t row = gid / N;
    float x0 = X[row * 2], x1 = X[row * 2 + 1];
    float w0, w1;
    if (col < F) { w0 = W0[col];           w1 = W0[F + col]; }
    else         { w0 = W1[col - F];       w1 = W1[F + col - F]; }
    H[gid] = fmaf(x0, w0, x1 * w1);
}

// ---------------------------------------------------------------------------
// Chain-graph softmax coefficients (row-major nnz ordering j=i-1,i,i+1).
// coef[i]=diag, coef[32+i]=left, coef[64+i]=right; absent neighbors -> 0.
// ---------------------------------------------------------------------------
__global__ void coeff_kernel(const float* __restrict__ e,
                             float* __restrict__ coef, int n)
{
    int i = threadIdx.x;
    if (i >= n) return;
    int base = (i == 0) ? 0 : 3 * i - 1;
    bool hasL = (i > 0), hasR = (i < n - 1);
    float eL = hasL ? e[base] : -3.0e38f;
    float eD = e[base + (hasL ? 1 : 0)];
    float eR = hasR ? e[base + (hasL ? 1 : 0) + 1] : -3.0e38f;
    float m  = fmaxf(eD, fmaxf(eL, eR));
    float xD = __expf(eD - m);
    float xL = hasL ? __expf(eL - m) : 0.0f;
    float xR = hasR ? __expf(eR - m) : 0.0f;
    float s  = xD + xL + xR;
    coef[i]      = xD / s;
    coef[32 + i] = xL / s;
    coef[64 + i] = xR / s;
}

// ---------------------------------------------------------------------------
// Graph mix: out[b,i,f] = d[i]*h0[b,i,f] + aL[i]*h1[b,i-1,f]
//                       + aR[i]*h1[b,i+1,f] + bias[f]
// mode 0: store raw (pre-BN).  mode 1: sigmoid (output head).
// ---------------------------------------------------------------------------
__global__ __launch_bounds__(256) void mix_kernel(
    const float* __restrict__ H, const float* __restrict__ coef,
    const float* __restrict__ bias, float* __restrict__ out,
    int R, int F, int n, int mode)
{
    size_t gid = (size_t)blockIdx.x * 256 + threadIdx.x;
    if (gid >= (size_t)R * F) return;
    int row = (int)(gid / F);
    int f   = (int)(gid % F);
    int i   = row % n;
    const size_t N2 = (size_t)2 * F;
    float v = coef[i] * H[(size_t)row * N2 + f] + bias[f];
    if (i > 0)     v += coef[32 + i] * H[(size_t)(row - 1) * N2 + F + f];
    if (i < n - 1) v += coef[64 + i] * H[(size_t)(row + 1) * N2 + F + f];
    if (mode == 1) v = 1.0f / (1.0f + __expf(-v));
    out[gid] = v;
}

// ---------------------------------------------------------------------------
// BatchNorm stats: per-block partial column sums / sumsq (coalesced rows).
// ---------------------------------------------------------------------------
__global__ __launch_bounds__(256) void stats_partial(
    const float* __restrict__ P, float* __restrict__ ps,
    float* __restrict__ pss, int R, int F)
{
    int t = threadIdx.x, blk = blockIdx.x;
    int rows_per = (R + STATS_BLOCKS - 1) / STATS_BLOCKS;
    int r0 = blk * rows_per;
    int r1 = r0 + rows_per; if (r1 > R) r1 = R;
    for (int f = t; f < F; f += 256) {
        float s = 0.0f, ss = 0.0f;
        for (int r = r0; r < r1; r++) {
            float v = P[(size_t)r * F + f];
            s += v; ss += v * v;
        }
        ps[blk * F + f]  = s;
        pss[blk * F + f] = ss;
    }
}

__global__ __launch_bounds__(512) void bn_finalize(
    const float* __restrict__ ps, const float* __restrict__ pss,
    const float* __restrict__ g, const float* __restrict__ be,
    float* __restrict__ scale, float* __restrict__ shift, int F, int R)
{
    int f = threadIdx.x;
    if (f >= F) return;
    float s = 0.0f, ss = 0.0f;
    for (int b = 0; b < STATS_BLOCKS; b++) { s += ps[b * F + f]; ss += pss[b * F + f]; }
    float mean = s / (float)R;
    float var  = ss / (float)R - mean * mean;
    float sc   = g[f] * rsqrtf(var + 1e-5f);
    scale[f] = sc;
    shift[f] = be[f] - mean * sc;
}

// out = relu(P*scale+shift) (+ addend for the residual c = c + h)
__global__ __launch_bounds__(256) void bn_apply(
    const float* __restrict__ P, const float* __restrict__ scale,
    const float* __restrict__ shift, const float* __restrict__ addend,
    float* __restrict__ out, int R, int F)
{
    size_t gid = (size_t)blockIdx.x * 256 + threadIdx.x;
    if (gid >= (size_t)R * F) return;
    int f = (int)(gid % F);
    float v = fmaxf(P[gid] * scale[f] + shift[f], 0.0f);
    if (addend) v += addend[gid];
    out[gid] = v;
}

// ---------------------------------------------------------------------------
// CAT[b,n,0:160] = C[b,n,:]; CAT[b,n,160:320] = s_r (reshape/transpose of S).
// s_r[b,n,f'] with l = n*160+f' -> S[b, l%10, l/10]   (S is [B,10,512])
// ---------------------------------------------------------------------------
__global__ __launch_bounds__(256) void build_cat(
    const float* __restrict__ Cc, const float* __restrict__ Ss,
    float* __restrict__ CAT, int B)
{
    size_t gid = (size_t)blockIdx.x * 256 + threadIdx.x;
    size_t total = (size_t)B * 32 * 320;
    if (gid >= total) return;
    int f = (int)(gid % 320);
    size_t row = gid / 320;              // b*32 + n
    if (f < 160) {
        CAT[gid] = Cc[row * 160 + f];
    } else {
        int b  = (int)(row >> 5);
        int nn = (int)(row & 31);
        int l  = nn * 160 + (f - 160);
        int ch = l / 10, node = l % 10;
        CAT[gid] = Ss[((size_t)b * 10 + node) * 512 + ch];
    }
}

// ---------------------------------------------------------------------------
// Output head GEMM: H[R,4] = C[R,160] @ [W0 | W1], W0/W1 each [160,2].
// One wave per row, shuffle reduction (N=4 is too narrow for WMMA tiles).
// ---------------------------------------------------------------------------
__global__ __launch_bounds__(256) void gemm_outproj(
    const float* __restrict__ Cc, const float* __restrict__ W0,
    const float* __restrict__ W1, float* __restrict__ H, int R, int K)
{
    int gw   = (int)((blockIdx.x * (size_t)blockDim.x + threadIdx.x) >> 5);
    int lane = threadIdx.x & 31;
    if (gw >= R) return;
    const float* xr = Cc + (size_t)gw * K;
    float a0 = 0, a1 = 0, a2 = 0, a3 = 0;
    for (int k = lane; k < K; k += 32) {
        float x = xr[k];
        a0 += x * W0[k * 2 + 0];
        a1 += x * W0[k * 2 + 1];
        a2 += x * W1[k * 2 + 0];
        a3 += x * W1[k * 2 + 1];
    }
    #pragma unroll
    for (int o = 16; o > 0; o >>= 1) {
        a0 += __shfl_xor(a0, o, 32);
        a1 += __shfl_xor(a1, o, 32);
        a2 += __shfl_xor(a2, o, 32);
        a3 += __shfl_xor(a3, o, 32);
    }
    if (lane == 0) {
        H[(size_t)gw * 4 + 0] = a0;
        H[(size_t)gw * 4 + 1] = a1;
        H[(size_t)gw * 4 + 2] = a2;
        H[(size_t)gw * 4 + 3] = a3;
    }
}

// ---------------------------------------------------------------------------
// Host orchestration
// ---------------------------------------------------------------------------
namespace {

struct Wsp {
    float *coef, *scale, *shift, *ps, *pss;
    float *C, *T1, *S, *P, *CAT, *HB;
};

inline void run_gblock(hipStream_t stream, const Wsp& w,
                       const float* X, int R, int K, int F, int n,
                       const float* W0, const float* W1, const float* e,
                       const float* bias, const float* g, const float* be,
                       const float* addend, float* out)
{
    coeff_kernel<<<1, 32, 0, stream>>>(e, w.coef, n);
    if (K == 2) {
        size_t tot = (size_t)R * 2 * F;
        gemm_k2<<<(unsigned)((tot + 255) / 256), 256, 0, stream>>>(X, W0, W1,
                                                                   w.HB, R, F);
    } else {
        dim3 gg(2 * F / GBN, R / GBM);
        gemm_wmma<<<gg, 256, 0, stream>>>(X, W0, W1, w.HB, R, K, F);
    }
    int nel = R * F;
    mix_kernel<<<(nel + 255) / 256, 256, 0, stream>>>(w.HB, w.coef, bias, w.P,
                                                      R, F, n, 0);
    stats_partial<<<STATS_BLOCKS, 256, 0, stream>>>(w.P, w.ps, w.pss, R, F);
    bn_finalize<<<1, 512, 0, stream>>>(w.ps, w.pss, g, be, w.scale, w.shift, F, R);
    bn_apply<<<(nel + 255) / 256, 256, 0, stream>>>(w.P, w.scale, w.shift,
                                                    addend, out, R, F);
}

} // namespace

extern "C" void kernel_launch(void* const* d_in, const int* in_sizes, int n_in,
                              void* d_out, int out_size, void* d_ws, size_t ws_size,
                              hipStream_t stream)
{
    const float* x_c     = (const float*)d_in[0];
    const float* x_s     = (const float*)d_in[1];
    const float* W_in    = (const float*)d_in[2];
    const float* e_in    = (const float*)d_in[3];
    const float* b_in    = (const float*)d_in[4];
    const float* g_in    = (const float*)d_in[5];
    const float* be_in   = (const float*)d_in[6];
    const float* W_s1    = (const float*)d_in[7];
    const float* e_s1    = (const float*)d_in[8];
    const float* b_s1    = (const float*)d_in[9];
    const float* g_s1    = (const float*)d_in[10];
    const float* be_s1   = (const float*)d_in[11];
    const float* W_c320  = (const float*)d_in[12];
    const float* e_c320  = (const float*)d_in[13];
    const float* b_c320  = (const float*)d_in[14];
    const float* g_c320  = (const float*)d_in[15];
    const float* be_c320 = (const float*)d_in[16];
    const float* W_c160  = (const float*)d_in[17];
    const float* e_c160  = (const float*)d_in[18];
    const float* b_c160  = (const float*)d_in[19];
    const float* g_c160  = (const float*)d_in[20];
    const float* be_c160 = (const float*)d_in[21];
    const float* W_s512  = (const float*)d_in[22];
    const float* e_s512  = (const float*)d_in[23];
    const float* b_s512  = (const float*)d_in[24];
    const float* g_s512  = (const float*)d_in[25];
    const float* be_s512 = (const float*)d_in[26];
    const float* W_out   = (const float*)d_in[27];
    const float* e_out   = (const float*)d_in[28];
    const float* b_out   = (const float*)d_in[29];

    constexpr int B = 4096, NC = 32, NS = 10, Hf = 160, HS = 512;
    constexpr int RC = B * NC;          // 131072
    constexpr int RS = B * NS;          // 40960
    constexpr int NNZ_C = 3 * NC - 2;   // 94
    constexpr int NNZ_S = 3 * NS - 2;   // 28
    constexpr size_t SZ_C   = (size_t)RC * Hf;       // 20,971,520 floats
    constexpr size_t SZ_CAT = (size_t)RC * 2 * Hf;

    float* wsf = (float*)d_ws;
    Wsp w;
    w.coef  = wsf;              // 128
    w.scale = wsf + 128;        // 512
    w.shift = wsf + 640;        // 512
    w.ps    = wsf + 1152;       // 256*512
    w.pss   = w.ps + STATS_BLOCKS * 512;
    float* big = wsf + (1 << 19);
    w.C   = big;
    w.T1  = w.C  + SZ_C;
    w.S   = w.T1 + SZ_C;        // [RS, 512] == SZ_C floats
    w.P   = w.S  + SZ_C;
    w.CAT = w.P  + SZ_C;
    w.HB  = w.CAT + SZ_CAT;

    // c = gblock(x_c, W_in)      K=2 -> F=160
    run_gblock(stream, w, x_c, RC, 2, Hf, NC,
               W_in, W_in + 2 * Hf, e_in, b_in, g_in, be_in, nullptr, w.C);

    // s = gblock(x_s, W_s1)      K=2 -> F=512
    run_gblock(stream, w, x_s, RS, 2, HS, NS,
               W_s1, W_s1 + 2 * HS, e_s1, b_s1, g_s1, be_s1, nullptr, w.S);

    for (int i = 0; i < 3; i++) {
        // CAT = [c | reshape(transpose(s))]
        size_t tot = (size_t)RC * 2 * Hf;
        build_cat<<<(unsigned)((tot + 255) / 256), 256, 0, stream>>>(w.C, w.S,
                                                                     w.CAT, B);

        // c = gblock(CAT, W_c320[i])   K=320 -> F=160
        const float* Wc = W_c320 + (size_t)i * 2 * 320 * Hf;
        run_gblock(stream, w, w.CAT, RC, 2 * Hf, Hf, NC,
                   Wc, Wc + 320 * Hf, e_c320 + i * NNZ_C,
                   b_c320 + i * Hf, g_c320 + i * Hf, be_c320 + i * Hf,
                   nullptr, w.C);

        // h = gblock(c, W_c160[2i])    K=160 -> F=160
        const float* Wh0 = W_c160 + (size_t)(2 * i) * 2 * Hf * Hf;
        run_gblock(stream, w, w.C, RC, Hf, Hf, NC,
                   Wh0, Wh0 + Hf * Hf, e_c160 + (2 * i) * NNZ_C,
                   b_c160 + (2 * i) * Hf, g_c160 + (2 * i) * Hf,
                   be_c160 + (2 * i) * Hf, nullptr, w.T1);

        // h = gblock(h, W_c160[2i+1]); c = c + h  (residual fused into apply)
        const float* Wh1 = W_c160 + (size_t)(2 * i + 1) * 2 * Hf * Hf;
        run_gblock(stream, w, w.T1, RC, Hf, Hf, NC,
                   Wh1, Wh1 + Hf * Hf, e_c160 + (2 * i + 1) * NNZ_C,
                   b_c160 + (2 * i + 1) * Hf, g_c160 + (2 * i + 1) * Hf,
                   be_c160 + (2 * i + 1) * Hf, w.C, w.C);

        if (i < 2) {
            // s = gblock(s, W_s512[i])  K=512 -> F=512
            const float* Wsq = W_s512 + (size_t)i * 2 * HS * HS;
            run_gblock(stream, w, w.S, RS, HS, HS, NS,
                       Wsq, Wsq + HS * HS, e_s512 + i * NNZ_S,
                       b_s512 + i * HS, g_s512 + i * HS, be_s512 + i * HS,
                       nullptr, w.S);
        }
    }

    // Output head: semgconv(c, W_out) -> sigmoid into d_out [B,32,2]
    coeff_kernel<<<1, 32, 0, stream>>>(e_out, w.coef, NC);
    gemm_outproj<<<RC / 8, 256, 0, stream>>>(w.C, W_out, W_out + Hf * 2,
                                             w.HB, RC, Hf);
    int nel_out = RC * 2;
    mix_kernel<<<(nel_out + 255) / 256, 256, 0, stream>>>(
        w.HB, w.coef, b_out, (float*)d_out, RC, 2, NC, 1);
}